// Gaussian_77859167142553
// MI455X (gfx1250) — compile-verified
//
#include <hip/hip_runtime.h>
#include <math.h>

typedef __attribute__((ext_vector_type(2))) float v2f;
typedef __attribute__((ext_vector_type(4))) float v4f;
typedef __attribute__((ext_vector_type(8))) float v8f;

#define DD 128
#define TPW 4  // point-tiles per wave

// ---------------------------------------------------------------------------
// Kernel 1: S = sigma + eps; Cholesky S = L L^T (lower, in LDS);
//           W = L^{-1} (lower triangular) -> workspace; log_den -> workspace.
// One workgroup, 128 threads. O(D^3) ~ negligible.
// ---------------------------------------------------------------------------
__global__ __launch_bounds__(128)
void chol_inv_kernel(const float* __restrict__ sigma,
                     const float* __restrict__ eps,
                     float* __restrict__ Wout,
                     float* __restrict__ logden_out)
{
    __shared__ float sS[DD * DD];
    const int tid = threadIdx.x;

    for (int i = tid; i < DD * DD; i += 128) sS[i] = sigma[i] + eps[i];
    __syncthreads();

    // In-place Cholesky (lower). Thread t owns row t.
    for (int k = 0; k < DD; ++k) {
        float dkk = sqrtf(sS[k * DD + k]);   // everyone reads pre-update diag
        __syncthreads();
        if (tid == k) sS[k * DD + k] = dkk;
        if (tid > k)  sS[tid * DD + k] /= dkk;
        __syncthreads();
        if (tid > k) {
            float lik = sS[tid * DD + k];
            for (int j = k + 1; j <= tid; ++j)
                sS[tid * DD + j] -= lik * sS[j * DD + k];
        }
        __syncthreads();
    }

    // Forward substitution: thread c solves L w = e_c  ->  column c of W=L^{-1}
    {
        const int c = tid;
        float w[DD];
        for (int r = 0; r < DD; ++r) w[r] = 0.0f;
        w[c] = 1.0f / sS[c * DD + c];
        for (int r = c + 1; r < DD; ++r) {
            float s = 0.0f;
            for (int j = c; j < r; ++j) s += sS[r * DD + j] * w[j];
            w[r] = -s / sS[r * DD + r];
        }
        for (int r = 0; r < DD; ++r)
            Wout[r * DD + c] = (r >= c) ? w[r] : 0.0f;
    }

    if (tid == 0) {
        float logdet = 0.0f;
        for (int k = 0; k < DD; ++k) logdet += 2.0f * logf(sS[k * DD + k]);
        const float log2pi = logf(6.28318530717958647692f);
        *logden_out = 0.5f * (128.0f * log2pi + logdet);
    }
}

// ---------------------------------------------------------------------------
// Kernel 2: per point n, quad = || W * (x_n - mu) ||^2 via fp32 WMMA,
//           out[n] = 0.5*quad + log_den.
//
// K-slot permutation: dim group 8g..8g+7 feeds WMMA steps 2g and 2g+1:
//   step 2g  : lanes 0-15 dims 8g+0,8g+1 ; lanes 16-31 dims 8g+4,8g+5
//   step 2g+1: lanes 0-15 dims 8g+2,8g+3 ; lanes 16-31 dims 8g+6,8g+7
// -> each lane loads one b128 (4 consecutive dims) per step pair.
// W lower-triangular: row-block rb only needs steps 0..4rb+3 (144/256 tiles).
// ---------------------------------------------------------------------------
__global__ __launch_bounds__(256)
void gauss_nll_kernel(const float* __restrict__ X,
                      const float* __restrict__ mu,
                      const float* __restrict__ W,
                      const float* __restrict__ logden_p,
                      float* __restrict__ out,
                      int Nn)
{
    __shared__ float sW[DD * DD];
    __shared__ float sMu[DD];

    for (int i = threadIdx.x; i < DD * DD; i += 256) sW[i] = W[i];
    if (threadIdx.x < DD) sMu[threadIdx.x] = mu[threadIdx.x];
    __syncthreads();

    const float logden = *logden_p;
    const int wave = threadIdx.x >> 5;
    const int lane = threadIdx.x & 31;
    const int half = lane >> 4;     // 0: lanes 0-15, 1: lanes 16-31
    const int l16  = lane & 15;

#pragma unroll 1
    for (int t = 0; t < TPW; ++t) {
        // Compiler barrier: forbid hoisting/spilling of per-tile LDS reads
        // (W A-fragments, mu) across tile iterations.
        asm volatile("" ::: "memory");

        const int tile  = (blockIdx.x * 8 + wave) * TPW + t;
        const int point = tile * 16 + l16;
        const int p     = point < Nn ? point : Nn - 1;   // keep EXEC full
        const float* xrow = X + (size_t)p * DD;

        // Prefetch next tile's X row (global_prefetch); this wave's next
        // iteration reads xrow + 16 rows = +8KB.
        __builtin_prefetch(xrow + 16 * DD + 64 * half, 0, 0);

        // Whitened-diff B data: 16 x b128 streaming loads per lane.
        v4f xb[16];
#pragma unroll
        for (int g = 0; g < 16; ++g) {
            const int c = 8 * g + 4 * half;              // 4 consecutive dims
            v4f xv = __builtin_nontemporal_load((const v4f*)(xrow + c));
            v4f mv = *(const v4f*)(&sMu[c]);
            xb[g] = xv - mv;
        }

        float q = 0.0f;
#pragma unroll
        for (int rb = 0; rb < 8; ++rb) {
            v8f acc = {0.f, 0.f, 0.f, 0.f, 0.f, 0.f, 0.f, 0.f};
            const int arow = 16 * rb + l16;
#pragma unroll
            for (int s = 0; s < 4 * rb + 4; ++s) {       // triangular skip
                const int g = s >> 1, odd = s & 1;
                v2f bb = odd ? __builtin_shufflevector(xb[g], xb[g], 2, 3)
                             : __builtin_shufflevector(xb[g], xb[g], 0, 1);
                v2f a = *(const v2f*)(&sW[arow * DD + 8 * g + 4 * half + 2 * odd]);
                acc = __builtin_amdgcn_wmma_f32_16x16x4_f32(
                        false, a, false, bb, (short)0, acc, false, false);
            }
#pragma unroll
            for (int r = 0; r < 8; ++r) q += acc[r] * acc[r];
        }

        // lanes 0-15 hold rows 0-7 of each block, lanes 16-31 rows 8-15:
        q += __shfl_xor(q, 16);
        if (half == 0 && point < Nn)
            __builtin_nontemporal_store(0.5f * q + logden, &out[point]);
    }
}

// ---------------------------------------------------------------------------
extern "C" void kernel_launch(void* const* d_in, const int* in_sizes, int n_in,
                              void* d_out, int out_size, void* d_ws, size_t ws_size,
                              hipStream_t stream)
{
    (void)n_in; (void)out_size; (void)ws_size;
    const float* X     = (const float*)d_in[0];
    const float* mu    = (const float*)d_in[1];
    const float* sigma = (const float*)d_in[2];
    const float* eps   = (const float*)d_in[3];
    float* out = (float*)d_out;

    float* wsf    = (float*)d_ws;
    float* Wm     = wsf;             // 128*128 floats
    float* logden = wsf + DD * DD;   // 1 float

    const int Nn = in_sizes[0] / DD;

    chol_inv_kernel<<<1, 128, 0, stream>>>(sigma, eps, Wm, logden);

    const int pointsPerBlock = 8 * TPW * 16;  // 512
    const int blocks = (Nn + pointsPerBlock - 1) / pointsPerBlock;
    gauss_nll_kernel<<<blocks, 256, 0, stream>>>(X, mu, Wm, logden, out, Nn);
}